// Build_Sinenet_V0_12438225289941
// MI455X (gfx1250) — compile-verified
//
#include <hip/hip_runtime.h>
#include <hip/hip_bf16.h>
#include <math.h>

// ---- problem constants (from reference) ----
#define NUM_FREQ   32
#define TWO_PI_K   6.283185307179586f   // 2*pi * k_space (k_space = 1.0)
#define WIN_LEN    640
#define T_WAV      (1.0f / 24000.0f)
#define D_OUT      512
#define LOG_F_MEAN 5.04418f
#define LOG_F_STD  0.358402f
#define N_ROWS     (4 * 16 * 40)        // S*B*M = 2560
#define K_AUG      68                   // 64 sin/cos + nlf + tau + vuv + 1(bias)
#define K_PAIRS    (K_AUG / 2)          // 34 row-pairs in packed B

typedef __attribute__((ext_vector_type(2))) float v2f;
typedef __attribute__((ext_vector_type(8))) float v8f;

// ---------------------------------------------------------------------------
// Prep: pack augmented B matrix, PAIR-MAJOR, into workspace:
//   Bpair[p][col] = ( Brow[2p][col], Brow[2p+1][col] )   as float2
// where Brow is the augmented [68][512] matrix:
//   rows 0..63  = W1, rows 64..66 = W2, row 67 = b1 + b2.
// This makes every WMMA B-fragment a single 8-byte global_load_b64.
// ---------------------------------------------------------------------------
__device__ __forceinline__ float baug_row(const float* __restrict__ W1,
                                          const float* __restrict__ b1,
                                          const float* __restrict__ W2,
                                          const float* __restrict__ b2,
                                          int k, int d)
{
    if (k < 64) return W1[k * D_OUT + d];
    if (k < 67) return W2[(k - 64) * D_OUT + d];
    return b1[d] + b2[d];
}

__global__ __launch_bounds__(256) void sinenet_prep_bpair(
    const float* __restrict__ W1, const float* __restrict__ b1,
    const float* __restrict__ W2, const float* __restrict__ b2,
    float2* __restrict__ Bpair)
{
    int idx = blockIdx.x * 256 + threadIdx.x;      // 0 .. 34*512-1
    if (idx >= K_PAIRS * D_OUT) return;
    int p = idx >> 9;                              // pair index 0..33
    int d = idx & (D_OUT - 1);
    float2 v;
    v.x = baug_row(W1, b1, W2, b2, 2 * p,     d);
    v.y = baug_row(W1, b1, W2, b2, 2 * p + 1, d);
    Bpair[idx] = v;
}

// ---------------------------------------------------------------------------
// Fused kernel: one block = 16 rows of (S,B,M). 256 threads = 8 wave32s.
//   Phase 0: stage wav tile [16][640] in LDS (float4 loads).
//   Phase 1: per-thread sincos reduction over T -> scx[16][68] in LDS.
//   Phase 2: WMMA f32 16x16x4: [16 x 68] @ [68 x 512] (+bias folded), ReLU.
// ---------------------------------------------------------------------------
__global__ __launch_bounds__(256) void sinenet_fused(
    const float* __restrict__ wav,    // [2560][640]
    const float* __restrict__ nlf,    // [2560]
    const float* __restrict__ tau,    // [2560]
    const float* __restrict__ vuv,    // [2560]
    const float2* __restrict__ Bpair, // [34][512] pair-packed augmented B
    float* __restrict__ out)          // [2560][512]
{
    __shared__ __align__(16) float wavTile[16][WIN_LEN];  // 40 KB
    __shared__ float scx[16][K_AUG];                      // stride 68: conflict-free A reads

    const int tid     = threadIdx.x;
    const int rowBase = blockIdx.x * 16;

    // ---- Phase 0: cooperative tile load (16*640 floats = 2560 float4) ----
    {
        const float4* gw = (const float4*)(wav + (size_t)rowBase * WIN_LEN);
        float4*       sw = (float4*)(&wavTile[0][0]);
        #pragma unroll
        for (int i = 0; i < 10; ++i)
            sw[tid + i * 256] = gw[tid + i * 256];
    }
    __syncthreads();

    // ---- Phase 1: sin/cos projection. Thread owns (row0,k) and (row0+8,k) ----
    {
        const int k    = tid & 31;       // frequency index 0..31
        const int row0 = tid >> 5;       // 0..7
        const int row1 = row0 + 8;

        const float kf  = (float)(k + 1) * TWO_PI_K;
        const float f0  = __expf(nlf[rowBase + row0] * LOG_F_STD + LOG_F_MEAN);
        const float f1  = __expf(nlf[rowBase + row1] * LOG_F_STD + LOG_F_MEAN);
        const float om0 = kf * f0;
        const float om1 = kf * f1;
        const float bb0 = -om0 * tau[rowBase + row0];
        const float bb1 = -om1 * tau[rowBase + row1];
        const float st0 = om0 * T_WAV;
        const float st1 = om1 * T_WAV;

        float as0 = 0.f, ac0 = 0.f, as1 = 0.f, ac1 = 0.f;
        #pragma unroll 4
        for (int t = 0; t < WIN_LEN; ++t) {
            const float ft = (float)t;
            const float w0 = wavTile[row0][t];    // full-wave broadcast read
            const float w1 = wavTile[row1][t];
            const float x0 = fmaf(ft, st0, bb0);
            const float x1 = fmaf(ft, st1, bb1);
            as0 = fmaf(__sinf(x0), w0, as0);
            ac0 = fmaf(__cosf(x0), w0, ac0);
            as1 = fmaf(__sinf(x1), w1, as1);
            ac1 = fmaf(__cosf(x1), w1, ac1);
        }
        scx[row0][k]            = as0;   // concat(sin, cos): sin -> 0..31
        scx[row0][NUM_FREQ + k] = ac0;   //                   cos -> 32..63
        scx[row1][k]            = as1;
        scx[row1][NUM_FREQ + k] = ac1;

        if (tid < 16) {                  // augmented columns
            const int g = rowBase + tid;
            scx[tid][64] = nlf[g];
            scx[tid][65] = tau[g];
            scx[tid][66] = vuv[g];
            scx[tid][67] = 1.0f;
        }
    }
    __syncthreads();

    // ---- Phase 2: WMMA f32 16x16x4. Wave w covers cols [w*64, w*64+64) ----
    {
        const int wave = tid >> 5;
        const int lane = tid & 31;
        const int kh   = lane >> 4;      // K-half select (ISA 16x4 f32 A layout)
        const int lr   = lane & 15;      // M row for A, N col for B/D

        #pragma unroll
        for (int nt = 0; nt < 4; ++nt) {
            const int col = wave * 64 + nt * 16 + lr;
            v8f c = {};
            #pragma unroll
            for (int kt = 0; kt < 17; ++kt) {        // K = 68 = 17 * 4
                const int k0 = kt * 4 + 2 * kh;      // = 2 * (kt*2 + kh)
                const int p  = kt * 2 + kh;          // packed pair row
                v2f a, b;
                a.x = scx[lr][k0];                   // A[M=lr][k0]
                a.y = scx[lr][k0 + 1];               // A[M=lr][k0+1]
                const float2 bv = Bpair[(size_t)p * D_OUT + col];  // one b64 load
                b.x = bv.x;                          // B[k0][col]
                b.y = bv.y;                          // B[k0+1][col]
                c = __builtin_amdgcn_wmma_f32_16x16x4_f32(
                        /*neg_a=*/false, a, /*neg_b=*/false, b,
                        /*c_mod=*/(short)0, c,
                        /*reuse_a=*/false, /*reuse_b=*/false);
            }
            // D layout: VGPR i -> M=i (lanes 0-15), M=i+8 (lanes 16-31)
            #pragma unroll
            for (int i = 0; i < 8; ++i) {
                const int m = i + 8 * kh;
                out[(size_t)(rowBase + m) * D_OUT + col] = fmaxf(c[i], 0.0f);
            }
        }
    }
}

// ---------------------------------------------------------------------------
extern "C" void kernel_launch(void* const* d_in, const int* in_sizes, int n_in,
                              void* d_out, int out_size, void* d_ws, size_t ws_size,
                              hipStream_t stream) {
    const float* wav = (const float*)d_in[0];   // [4,16,40,640]
    const float* nlf = (const float*)d_in[1];   // [4,16,40]
    const float* tau = (const float*)d_in[2];   // [4,16,40]
    const float* vuv = (const float*)d_in[3];   // [4,16,40]
    const float* W1  = (const float*)d_in[4];   // [64,512]
    const float* b1  = (const float*)d_in[5];   // [512]
    const float* W2  = (const float*)d_in[6];   // [3,512]
    const float* b2  = (const float*)d_in[7];   // [512]
    float*       out = (float*)d_out;           // [2560,512]
    float2*      Bpair = (float2*)d_ws;         // 34*512*8 = 136 KB scratch

    // Pack augmented B (W1 | W2 | b1+b2) pair-major -> workspace.
    sinenet_prep_bpair<<<(K_PAIRS * D_OUT + 255) / 256, 256, 0, stream>>>(
        W1, b1, W2, b2, Bpair);

    // Fused sincos projection + WMMA GEMM + ReLU.
    sinenet_fused<<<N_ROWS / 16, 256, 0, stream>>>(
        wav, nlf, tau, vuv, Bpair, out);
}